// mamba_SS_model_2430951490063
// MI455X (gfx1250) — compile-verified
//
#include <hip/hip_runtime.h>
#include <math.h>

// Problem constants (from reference)
#define D_MODEL   192
#define D_INNER   384
#define D_STATE   16
#define DT_RANK   12
#define BATCH     32
#define SEQLEN    2048
#define M_TOT     (BATCH * SEQLEN)   // 65536 rows
#define XDBL_PAD  48                 // DT_RANK + 2*D_STATE = 44, padded to 48
#define LN_EPS    1e-5f

typedef float v2f __attribute__((ext_vector_type(2)));
typedef float v8f __attribute__((ext_vector_type(8)));

// ---------------------------------------------------------------------------
// WMMA f32 16x16x4 fragment layouts (ISA 7.12.2):
//   A-frag: lane%16 = M row; VGPR0/1 hold K = (lane/16)*2 + {0,1}
//   B-frag: lane%16 = N col; VGPR0/1 hold K = (lane/16)*2 + {0,1}
//   C/D   : VGPR r -> row m0 + r + (lane>=16 ? 8 : 0), col = n0 + lane%16
// ---------------------------------------------------------------------------

// 2x4 register-blocked tile accumulator: wave computes a 32x64 output patch.
// Per k-step: 2 A-frag loads + 4 B-frag loads -> 8 WMMAs (load:wmma = 0.75).
__device__ __forceinline__ void wmma_tile_2x4(const float* __restrict__ A, int lda,
                                              const float* __restrict__ W, int ldw,
                                              int m0, int n0, int K, int lane,
                                              v8f c[2][4]) {
#pragma unroll
  for (int i = 0; i < 2; ++i)
#pragma unroll
    for (int j = 0; j < 4; ++j)
      c[i][j] = (v8f){0.f, 0.f, 0.f, 0.f, 0.f, 0.f, 0.f, 0.f};

  const int fr = lane & 15;
  const int kk = (lane >> 4) << 1;
  const float* __restrict__ Ap0 = A + (size_t)(m0 + fr) * lda + kk;
  const float* __restrict__ Ap1 = Ap0 + (size_t)16 * lda;
  const float* __restrict__ Wp0 = W + (size_t)(n0 + fr) * ldw + kk;
  const float* __restrict__ Wp1 = Wp0 + (size_t)16 * ldw;
  const float* __restrict__ Wp2 = Wp0 + (size_t)32 * ldw;
  const float* __restrict__ Wp3 = Wp0 + (size_t)48 * ldw;

  for (int k = 0; k < K; k += 4) {
    v2f a0, a1, b0, b1, b2, b3;
    a0[0] = Ap0[k]; a0[1] = Ap0[k + 1];
    a1[0] = Ap1[k]; a1[1] = Ap1[k + 1];
    b0[0] = Wp0[k]; b0[1] = Wp0[k + 1];
    b1[0] = Wp1[k]; b1[1] = Wp1[k + 1];
    b2[0] = Wp2[k]; b2[1] = Wp2[k + 1];
    b3[0] = Wp3[k]; b3[1] = Wp3[k + 1];
    c[0][0] = __builtin_amdgcn_wmma_f32_16x16x4_f32(false, a0, false, b0, (short)0, c[0][0], false, false);
    c[0][1] = __builtin_amdgcn_wmma_f32_16x16x4_f32(false, a0, false, b1, (short)0, c[0][1], false, false);
    c[0][2] = __builtin_amdgcn_wmma_f32_16x16x4_f32(false, a0, false, b2, (short)0, c[0][2], false, false);
    c[0][3] = __builtin_amdgcn_wmma_f32_16x16x4_f32(false, a0, false, b3, (short)0, c[0][3], false, false);
    c[1][0] = __builtin_amdgcn_wmma_f32_16x16x4_f32(false, a1, false, b0, (short)0, c[1][0], false, false);
    c[1][1] = __builtin_amdgcn_wmma_f32_16x16x4_f32(false, a1, false, b1, (short)0, c[1][1], false, false);
    c[1][2] = __builtin_amdgcn_wmma_f32_16x16x4_f32(false, a1, false, b2, (short)0, c[1][2], false, false);
    c[1][3] = __builtin_amdgcn_wmma_f32_16x16x4_f32(false, a1, false, b3, (short)0, c[1][3], false, false);
  }
}

// ---------------------------------------------------------------------------
// Plain blocked GEMM: C[M x N] = A[M x K] @ W[N x K]^T
// block = 256 (8 waves); wave w covers m-tiles blockIdx.y*16 + 2w + {0,1};
// grid.x covers N in 64-wide super-tiles. Requires N % 64 == 0, M % 256 == 0.
// ---------------------------------------------------------------------------
__global__ void gemm_wmma_2x4(const float* __restrict__ A, int lda,
                              const float* __restrict__ W, int ldw,
                              float* __restrict__ C, int ldc, int K) {
  const int lane = threadIdx.x & 31;
  const int wave = threadIdx.x >> 5;
  const int m0 = (blockIdx.y * 16 + wave * 2) * 16;
  const int n0 = blockIdx.x * 64;
  v8f c[2][4];
  wmma_tile_2x4(A, lda, W, ldw, m0, n0, K, lane, c);
  const int fr = lane & 15;
  const int rsel = (lane >> 4) << 3;
#pragma unroll
  for (int i = 0; i < 2; ++i) {
    const int row0 = m0 + i * 16 + rsel;
#pragma unroll
    for (int j = 0; j < 4; ++j) {
      const int col = n0 + j * 16 + fr;
#pragma unroll
      for (int r = 0; r < 8; ++r)
        C[(size_t)(row0 + r) * ldc + col] = c[i][j][r];
    }
  }
}

// ---------------------------------------------------------------------------
// dt GEMM (K = DT_RANK = 12 -> 3 WMMA k-steps) with bias + softplus epilogue:
// delta = softplus(dt_low @ W_dt^T + b_dt).  N = 384 (six 64-wide tiles).
// ---------------------------------------------------------------------------
__global__ void gemm_wmma_dt_2x4(const float* __restrict__ A, int lda,
                                 const float* __restrict__ W, int ldw,
                                 const float* __restrict__ bias,
                                 float* __restrict__ C, int ldc) {
  const int lane = threadIdx.x & 31;
  const int wave = threadIdx.x >> 5;
  const int m0 = (blockIdx.y * 16 + wave * 2) * 16;
  const int n0 = blockIdx.x * 64;
  v8f c[2][4];
  wmma_tile_2x4(A, lda, W, ldw, m0, n0, DT_RANK, lane, c);
  const int fr = lane & 15;
  const int rsel = (lane >> 4) << 3;
#pragma unroll
  for (int i = 0; i < 2; ++i) {
    const int row0 = m0 + i * 16 + rsel;
#pragma unroll
    for (int j = 0; j < 4; ++j) {
      const int col = n0 + j * 16 + fr;
      const float bd = bias[col];
#pragma unroll
      for (int r = 0; r < 8; ++r) {
        float v = c[i][j][r] + bd;
        float sp = (v > 20.f) ? v : log1pf(__expf(v));   // softplus
        C[(size_t)(row0 + r) * ldc + col] = sp;
      }
    }
  }
}

// ---------------------------------------------------------------------------
// x_proj GEMM, single 16x16 tile per wave, N padded to 48 (valid: 44).
// ---------------------------------------------------------------------------
__global__ void gemm_wmma_xproj(const float* __restrict__ A, int lda,
                                const float* __restrict__ W, int ldw,
                                float* __restrict__ C, int ldc, int K) {
  const int lane = threadIdx.x & 31;
  const int wave = threadIdx.x >> 5;
  const int m0 = (blockIdx.y * 8 + wave) * 16;
  const int n0 = blockIdx.x * 16;
  const int col = n0 + (lane & 15);
  const bool nv = (col < (DT_RANK + 2 * D_STATE));  // 44 valid output cols

  v8f c = {0.f, 0.f, 0.f, 0.f, 0.f, 0.f, 0.f, 0.f};
  const int kk = (lane >> 4) << 1;
  const float* __restrict__ Ap = A + (size_t)(m0 + (lane & 15)) * lda + kk;
  const float* __restrict__ Wp = W + (size_t)col * ldw + kk;
  for (int k = 0; k < K; k += 4) {
    v2f a, b;
    a[0] = Ap[k]; a[1] = Ap[k + 1];
    if (nv) { b[0] = Wp[k]; b[1] = Wp[k + 1]; }
    else    { b[0] = 0.f;   b[1] = 0.f; }
    c = __builtin_amdgcn_wmma_f32_16x16x4_f32(false, a, false, b,
                                              (short)0, c, false, false);
  }
  const int row0 = m0 + ((lane >> 4) << 3);
#pragma unroll
  for (int r = 0; r < 8; ++r)
    C[(size_t)(row0 + r) * ldc + col] = nv ? (float)c[r] : 0.f;  // pad cols = 0
}

// ---------------------------------------------------------------------------
// Selective scan: one thread per (b, d) lane; 16-state recurrence in VGPRs.
// Sequential over L (inherent); B/C rows broadcast from L2.
// ---------------------------------------------------------------------------
__global__ void scan_kernel(const float* __restrict__ delta,
                            const float* __restrict__ xz,     // u at stride 768
                            const float* __restrict__ xdbl,   // [M x 48]
                            const float* __restrict__ A_log,
                            const float* __restrict__ Dvec,
                            float* __restrict__ ypre) {
  const int gid = blockIdx.x * blockDim.x + threadIdx.x;
  const int b = gid / D_INNER;
  const int d = gid - b * D_INNER;

  float Arow[D_STATE];
#pragma unroll
  for (int n = 0; n < D_STATE; ++n)
    Arow[n] = -__expf(A_log[d * D_STATE + n]);

  float h[D_STATE];
#pragma unroll
  for (int n = 0; n < D_STATE; ++n) h[n] = 0.f;

  const float Dd = Dvec[d];
  const size_t rowBase = (size_t)b * SEQLEN;

  for (int l = 0; l < SEQLEN; ++l) {
    const size_t row = rowBase + l;
    const float dl = delta[row * D_INNER + d];
    const float ul = xz[row * (2 * D_INNER) + d];
    const float* __restrict__ bp = xdbl + row * XDBL_PAD + DT_RANK;
    float Bv[D_STATE], Cv[D_STATE];
#pragma unroll
    for (int n = 0; n < D_STATE; ++n) { Bv[n] = bp[n]; Cv[n] = bp[D_STATE + n]; }
    const float du = dl * ul;
    float y = 0.f;
#pragma unroll
    for (int n = 0; n < D_STATE; ++n) {
      const float dA = __expf(dl * Arow[n]);
      h[n] = fmaf(dA, h[n], du * Bv[n]);
      y = fmaf(h[n], Cv[n], y);
    }
    ypre[row * D_INNER + d] = fmaf(ul, Dd, y);
  }
}

// ---------------------------------------------------------------------------
// LayerNorm over D_INNER + SiLU gate. One wave32 per row (12 elems/lane).
// block = 256 (8 waves -> 8 rows); grid = M_TOT/8.
// ---------------------------------------------------------------------------
__global__ void ln_gate_kernel(const float* __restrict__ ypre,
                               const float* __restrict__ xz,   // z at +384, stride 768
                               const float* __restrict__ ln_w,
                               const float* __restrict__ ln_b,
                               float* __restrict__ yg) {
  const int lane = threadIdx.x & 31;
  const int wave = threadIdx.x >> 5;
  const int row  = blockIdx.x * 8 + wave;

  const float* __restrict__ yr = ypre + (size_t)row * D_INNER;
  const float* __restrict__ zr = xz + (size_t)row * (2 * D_INNER) + D_INNER;

  float v[12];
  float s = 0.f;
#pragma unroll
  for (int i = 0; i < 12; ++i) { v[i] = yr[lane + i * 32]; s += v[i]; }
#pragma unroll
  for (int off = 16; off > 0; off >>= 1) s += __shfl_xor(s, off, 32);
  const float mu = s * (1.f / D_INNER);

  float q = 0.f;
#pragma unroll
  for (int i = 0; i < 12; ++i) { float t = v[i] - mu; q = fmaf(t, t, q); }
#pragma unroll
  for (int off = 16; off > 0; off >>= 1) q += __shfl_xor(q, off, 32);
  const float rstd = rsqrtf(q * (1.f / D_INNER) + LN_EPS);

#pragma unroll
  for (int i = 0; i < 12; ++i) {
    const int idx = lane + i * 32;
    const float g = zr[idx];
    const float silu = g / (1.f + __expf(-g));
    const float yn = fmaf((v[i] - mu) * rstd, ln_w[idx], ln_b[idx]);
    yg[(size_t)row * D_INNER + idx] = yn * silu;
  }
}

// ---------------------------------------------------------------------------
// Host launcher
// ---------------------------------------------------------------------------
extern "C" void kernel_launch(void* const* d_in, const int* in_sizes, int n_in,
                              void* d_out, int out_size, void* d_ws, size_t ws_size,
                              hipStream_t stream) {
  (void)in_sizes; (void)n_in; (void)out_size; (void)ws_size;
  const float* x       = (const float*)d_in[0];
  const float* W_in    = (const float*)d_in[1];
  const float* W_xproj = (const float*)d_in[2];
  const float* W_dt    = (const float*)d_in[3];
  const float* b_dt    = (const float*)d_in[4];
  const float* A_log   = (const float*)d_in[5];
  const float* Dvec    = (const float*)d_in[6];
  const float* ln_w    = (const float*)d_in[7];
  const float* ln_b    = (const float*)d_in[8];
  const float* W_out   = (const float*)d_in[9];
  float* out = (float*)d_out;

  // Workspace layout (floats):
  //   xz    : M_TOT * 768   (u = xz[:, :384], z = xz[:, 384:])
  //   xdbl  : M_TOT * 48    (dt_low | B | C, padded 44->48)
  //   delta : M_TOT * 384   (reused as yg after scan)
  //   ypre  : M_TOT * 384
  float* ws    = (float*)d_ws;
  float* xz    = ws;
  float* xdbl  = xz   + (size_t)M_TOT * (2 * D_INNER);
  float* delta = xdbl + (size_t)M_TOT * XDBL_PAD;
  float* ypre  = delta + (size_t)M_TOT * D_INNER;
  float* yg    = delta;  // delta dead after scan

  const dim3 blk(256);
  const int mgrid2x4 = M_TOT / 256;  // 256 rows per block (8 waves x 2 m-tiles)

  // 1) in_proj: xz[M,768] = x[M,192] @ W_in[768,192]^T   (N=768 -> 12 super-tiles)
  gemm_wmma_2x4<<<dim3((2 * D_INNER) / 64, mgrid2x4), blk, 0, stream>>>(
      x, D_MODEL, W_in, D_MODEL, xz, 2 * D_INNER, D_MODEL);

  // 2) x_proj: xdbl[M,48] = u[M,384](stride 768) @ W_xproj[44,384]^T (padded)
  gemm_wmma_xproj<<<dim3(XDBL_PAD / 16, M_TOT / 128), blk, 0, stream>>>(
      xz, 2 * D_INNER, W_xproj, D_INNER, xdbl, XDBL_PAD, D_INNER);

  // 3) delta[M,384] = softplus(dt_low[M,12](stride 48) @ W_dt[384,12]^T + b_dt)
  gemm_wmma_dt_2x4<<<dim3(D_INNER / 64, mgrid2x4), blk, 0, stream>>>(
      xdbl, XDBL_PAD, W_dt, DT_RANK, b_dt, delta, D_INNER);

  // 4) selective scan -> ypre (includes u*D skip)
  scan_kernel<<<(BATCH * D_INNER) / 256, 256, 0, stream>>>(
      delta, xz, xdbl, A_log, Dvec, ypre);

  // 5) LayerNorm + SiLU gate -> yg (reuses delta buffer)
  ln_gate_kernel<<<M_TOT / 8, 256, 0, stream>>>(ypre, xz, ln_w, ln_b, yg);

  // 6) out_proj: out[M,192] = yg[M,384] @ W_out[192,384]^T  (N=192 -> 3 super-tiles)
  gemm_wmma_2x4<<<dim3(D_MODEL / 64, mgrid2x4), blk, 0, stream>>>(
      yg, D_INNER, W_out, D_INNER, out, D_MODEL, D_INNER);
}